// MinimalHPM_10007273800078
// MI455X (gfx1250) — compile-verified
//
#include <hip/hip_runtime.h>

typedef __attribute__((ext_vector_type(2))) float v2f;
typedef __attribute__((ext_vector_type(8))) float v8f;

#define LOG2E 1.4426950408889634f
#define C1 (-2.0f * LOG2E)   /* exp(-r2/(2*0.25)) -> exp2(C1*r2) */
#define C2 (-0.5f * LOG2E)   /* exp(-max(t,0)/2)  -> exp2(C2*mt) */

// q(z) = C1 * r2(z), quadratic in z for fixed (ix,iy,ray)
__device__ __forceinline__ float qeval(float z, float ix, float iy,
                                       float dx, float dy, float dz,
                                       float ox, float oy, float oz,
                                       float od, float d2m2) {
  float px = ix - ox, py = iy - oy, pz = z - oz;
  float t  = fmaf(ix, dx, fmaf(iy, dy, fmaf(z, dz, -od)));
  float r2 = fmaf(t * t, d2m2, fmaf(pz, pz, fmaf(py, py, px * px)));
  return C1 * r2;
}

__device__ __forceinline__ float teval(float z, float ix, float iy,
                                       float dx, float dy, float dz, float od) {
  return fmaf(ix, dx, fmaf(iy, dy, fmaf(z, dz, -od)));
}

__global__ void __launch_bounds__(256)
hpm_stage1(const float* __restrict__ rayO, const float* __restrict__ rayD,
           const float* __restrict__ mem, float* __restrict__ partials,
           int numWG) {
  const int tid  = threadIdx.x;
  const int lane = tid & 31;
  const int wv   = tid >> 5;
  const int col  = lane & 15;          // B/D column = channel; also A row = ray
  const int ka   = (lane >> 4) * 2;    // lane's K slots: ka, ka+1 (per ISA A layout)
  const int rL   = col;                // rays 0..15  -> A_lo
  const int rH   = col + 16;           // rays 16..31 -> A_hi

  // per-lane ray parameters (two rays per lane)
  const float oxL = rayO[rL * 3 + 0], oyL = rayO[rL * 3 + 1], ozL = rayO[rL * 3 + 2];
  const float dxL = rayD[rL * 3 + 0], dyL = rayD[rL * 3 + 1], dzL = rayD[rL * 3 + 2];
  const float oxH = rayO[rH * 3 + 0], oyH = rayO[rH * 3 + 1], ozH = rayO[rH * 3 + 2];
  const float dxH = rayD[rH * 3 + 0], dyH = rayD[rH * 3 + 1], dzH = rayD[rH * 3 + 2];
  const float odL = oxL * dxL + oyL * dyL + ozL * dzL;
  const float odH = oxH * dxH + oyH * dyH + ozH * dzH;
  const float d2m2L = fmaf(dxL, dxL, fmaf(dyL, dyL, dzL * dzL)) - 2.0f;
  const float d2m2H = fmaf(dxH, dxH, fmaf(dyH, dyH, dzH * dzH)) - 2.0f;
  const float dtL = 4.0f * dzL, dtH = 4.0f * dzH;

  v8f cLo = {0.f, 0.f, 0.f, 0.f, 0.f, 0.f, 0.f, 0.f};
  v8f cHi = {0.f, 0.f, 0.f, 0.f, 0.f, 0.f, 0.f, 0.f};

  const int totalWaves = numWG * 8;
  const int gw = blockIdx.x * 8 + wv;
  const float zA = (float)ka, zB = (float)(ka + 1);

  for (int line = gw; line < 128 * 128; line += totalWaves) {
    const float ix = (float)(line >> 7);
    const float iy = (float)(line & 127);
    const float* memLine = mem + (size_t)line * (128 * 16);

    // ---- init second-difference recurrences for q(z), step = 4 in z ----
    // ray L, slot a (z = ka + 4g) and slot b (z = ka+1 + 4g)
    float q0, q1, q2;
    q0 = qeval(zA,        ix, iy, dxL, dyL, dzL, oxL, oyL, ozL, odL, d2m2L);
    q1 = qeval(zA + 4.0f, ix, iy, dxL, dyL, dzL, oxL, oyL, ozL, odL, d2m2L);
    q2 = qeval(zA + 8.0f, ix, iy, dxL, dyL, dzL, oxL, oyL, ozL, odL, d2m2L);
    float qLa = q0, dqLa = q1 - q0;
    const float ddqL = (q2 - q1) - (q1 - q0);
    q0 = qeval(zB,        ix, iy, dxL, dyL, dzL, oxL, oyL, ozL, odL, d2m2L);
    q1 = qeval(zB + 4.0f, ix, iy, dxL, dyL, dzL, oxL, oyL, ozL, odL, d2m2L);
    float qLb = q0, dqLb = q1 - q0;
    float tLa = teval(zA, ix, iy, dxL, dyL, dzL, odL);
    float tLb = teval(zB, ix, iy, dxL, dyL, dzL, odL);
    // ray H
    q0 = qeval(zA,        ix, iy, dxH, dyH, dzH, oxH, oyH, ozH, odH, d2m2H);
    q1 = qeval(zA + 4.0f, ix, iy, dxH, dyH, dzH, oxH, oyH, ozH, odH, d2m2H);
    q2 = qeval(zA + 8.0f, ix, iy, dxH, dyH, dzH, oxH, oyH, ozH, odH, d2m2H);
    float qHa = q0, dqHa = q1 - q0;
    const float ddqH = (q2 - q1) - (q1 - q0);
    q0 = qeval(zB,        ix, iy, dxH, dyH, dzH, oxH, oyH, ozH, odH, d2m2H);
    q1 = qeval(zB + 4.0f, ix, iy, dxH, dyH, dzH, oxH, oyH, ozH, odH, d2m2H);
    float qHb = q0, dqHb = q1 - q0;
    float tHa = teval(zA, ix, iy, dxH, dyH, dzH, odH);
    float tHb = teval(zB, ix, iy, dxH, dyH, dzH, odH);

    // B tile pointer: mem[(line*128 + z)*16 + col], z = 4g + ka (+0/+1)
    const float* bp = memLine + (ka * 16 + col);

#pragma unroll 4
    for (int g = 0; g < 32; ++g) {
      const float b0 = bp[0];
      const float b1 = bp[16];
      bp += 64;

      v2f a_lo, a_hi, bm;
      a_lo.x = __builtin_amdgcn_exp2f(fmaf(fmaxf(tLa, 0.0f), C2, qLa));
      qLa += dqLa; dqLa += ddqL; tLa += dtL;
      a_lo.y = __builtin_amdgcn_exp2f(fmaf(fmaxf(tLb, 0.0f), C2, qLb));
      qLb += dqLb; dqLb += ddqL; tLb += dtL;
      a_hi.x = __builtin_amdgcn_exp2f(fmaf(fmaxf(tHa, 0.0f), C2, qHa));
      qHa += dqHa; dqHa += ddqH; tHa += dtH;
      a_hi.y = __builtin_amdgcn_exp2f(fmaf(fmaxf(tHb, 0.0f), C2, qHb));
      qHb += dqHb; dqHb += ddqH; tHb += dtH;
      bm.x = b0; bm.y = b1;

      // D(16x16) += A(16x4) x B(4x16): M=ray, N=channel, K=4 voxels along z
      cLo = __builtin_amdgcn_wmma_f32_16x16x4_f32(
          false, a_lo, false, bm, (short)0, cLo, false, false);
      cHi = __builtin_amdgcn_wmma_f32_16x16x4_f32(
          false, a_hi, false, bm, (short)0, cHi, false, false);
    }
  }

  // ---- reduce 8 wave-accumulators -> one 32x16 partial per workgroup ----
  __shared__ float buf[8][512];
  const int rowBase = (lane < 16) ? 0 : 8;  // D layout: VGPR j -> M = j or j+8
#pragma unroll
  for (int j = 0; j < 8; ++j) {
    buf[wv][(rowBase + j) * 16 + col]        = cLo[j];
    buf[wv][(16 + rowBase + j) * 16 + col]   = cHi[j];
  }
  __syncthreads();

  float s0 = 0.0f, s1 = 0.0f;
#pragma unroll
  for (int w = 0; w < 8; ++w) {
    s0 += buf[w][tid];
    s1 += buf[w][tid + 256];
  }
  partials[(size_t)blockIdx.x * 512 + tid]       = s0;
  partials[(size_t)blockIdx.x * 512 + tid + 256] = s1;
}

__global__ void __launch_bounds__(256)
hpm_stage2(const float* __restrict__ partials, float* __restrict__ out,
           int numWG) {
  const int o = blockIdx.x * 256 + threadIdx.x;  // 512 outputs, 2 blocks
  float s = 0.0f;
#pragma unroll 8
  for (int p = 0; p < numWG; ++p) s += partials[(size_t)p * 512 + o];
  out[o] = s;
}

extern "C" void kernel_launch(void* const* d_in, const int* in_sizes, int n_in,
                              void* d_out, int out_size, void* d_ws, size_t ws_size,
                              hipStream_t stream) {
  const float* rayO = (const float*)d_in[0];  // [32,3]
  const float* rayD = (const float*)d_in[1];  // [32,3]
  const float* mem  = (const float*)d_in[2];  // [128,128,128,16]
  float* out        = (float*)d_out;          // [32,16]
  float* partials   = (float*)d_ws;

  int numWG = 512;  // 4096 waves; 16384 z-lines -> 4 lines/wave
  const size_t bytesPerWG = 512 * sizeof(float);
  if ((size_t)numWG * bytesPerWG > ws_size) {
    numWG = (int)(ws_size / bytesPerWG);
    if (numWG < 1) numWG = 1;
  }

  hipLaunchKernelGGL(hpm_stage1, dim3(numWG), dim3(256), 0, stream,
                     rayO, rayD, mem, partials, numWG);
  hipLaunchKernelGGL(hpm_stage2, dim3(2), dim3(256), 0, stream,
                     partials, out, numWG);
}